// BiStochastic_59914793779439
// MI455X (gfx1250) — compile-verified
//
#include <hip/hip_runtime.h>

// CDNA5 / gfx1250 Sinkhorn (bi-stochastic normalization), factorized form:
//   out[r,j] = s0[r,j] * rowScale[r] * colScale[j]
// with scale vectors updated by 10 alternating matrix-vector reductions
// against the ORIGINAL matrix (kept L2-resident; 1 MB per batch).
// Column-pass reductions run on the matrix core via V_WMMA_F32_16X16X4_F32.

typedef __attribute__((ext_vector_type(2))) float v2f;
typedef __attribute__((ext_vector_type(8))) float v8f;

#define SK_N    512
#define SK_ITER 10
#define SK_EPS  1e-4f

__global__ __launch_bounds__(512)
void sinkhorn512_wmma(const float* __restrict__ S, float* __restrict__ out) {
  __shared__ float rowScale[SK_N];
  __shared__ float colScale[SK_N];

  const int b      = blockIdx.x;
  const float* A   = S   + (size_t)b * SK_N * SK_N;
  float*       O   = out + (size_t)b * SK_N * SK_N;

  const int tid  = threadIdx.x;     // 0..511
  const int lane = tid & 31;        // wave32
  const int wave = tid >> 5;        // 0..15

  rowScale[tid] = 1.0f;
  colScale[tid] = 1.0f;

  // Prime L2 for this batch tile: 1 MiB = 8192 cachelines, 16 per thread.
  {
    const char* p = (const char*)A;
    #pragma unroll
    for (int i = 0; i < 16; ++i)
      __builtin_prefetch(p + ((size_t)(tid + i * 512) << 7), 0, 1);
  }
  __syncthreads();

  const int half = lane >> 4;   // 0: lanes 0-15, 1: lanes 16-31
  const int n16  = lane & 15;   // column within 16-wide tile

  for (int it = 0; it < SK_ITER; ++it) {
    if ((it & 1) == 0) {
      // ---- Column pass on the matrix core ----------------------------------
      // T_j = sum_r A[r][j] * rowScale[r];  colScale[j] = 1/T_j
      // A-operand (16x4): every M-row holds w[r..r+3] -> D[m][n] = T partial.
      // Layout: VGPR0 = K=0 (lanes 0-15) / K=2 (lanes 16-31); VGPR1 = K=1 / K=3.
      // B-operand (4x16): VGPR0 = rows r+0 / r+2 ; VGPR1 = rows r+1 / r+3.
      for (int t = wave; t < SK_N / 16; t += 16) {   // 32 col-tiles, 2 per wave
        const int j0 = t * 16;
        v8f acc0 = {};
        v8f acc1 = {};
        for (int r = 0; r < SK_N; r += 8) {          // two K=4 chunks per step
          const int ra = r + 2 * half;               // chunk 0 rows for this half
          const int rb = ra + 4;                     // chunk 1 rows
          v2f wa, wb, ma, mb;
          wa.x = rowScale[ra];
          wa.y = rowScale[ra + 1];
          wb.x = rowScale[rb];
          wb.y = rowScale[rb + 1];
          const float* pa = A + (size_t)ra * SK_N + j0 + n16;
          const float* pb = pa + 4 * SK_N;
          ma.x = pa[0];                              // row ra   (coalesced 16-lane)
          ma.y = pa[SK_N];                           // row ra+1
          mb.x = pb[0];
          mb.y = pb[SK_N];
          acc0 = __builtin_amdgcn_wmma_f32_16x16x4_f32(
                   false, wa, false, ma, (short)0, acc0, false, false);
          acc1 = __builtin_amdgcn_wmma_f32_16x16x4_f32(
                   false, wb, false, mb, (short)0, acc1, false, false);
        }
        // All D rows are identical; lanes 0-15 of VGPR0 hold row M=0, N=0..15.
        const float T = acc0[0] + acc1[0];
        if (half == 0)
          colScale[j0 + n16] = 1.0f / T;
      }
    } else {
      // ---- Row pass: coalesced float4 loads + wave32 shuffle reduction -----
      // U_r = sum_j A[r][j]*colScale[j];  rowScale[r] = rs / (rs*U_r + eps)
      for (int r = wave; r < SK_N; r += 16) {
        const float* Ar = A + (size_t)r * SK_N;
        float acc = 0.0f;
        #pragma unroll
        for (int k = 0; k < SK_N / 128; ++k) {
          const int j = lane * 4 + k * 128;          // 16B-aligned
          const float4 av = *(const float4*)(Ar + j);
          const float4 cv = *(const float4*)(&colScale[j]);
          acc += av.x * cv.x + av.y * cv.y + av.z * cv.z + av.w * cv.w;
        }
        #pragma unroll
        for (int off = 16; off > 0; off >>= 1)
          acc += __shfl_down(acc, off, 32);
        if (lane == 0) {
          const float rs = rowScale[r];
          rowScale[r] = rs / (rs * acc + SK_EPS);
        }
      }
    }
    __syncthreads();
  }

  // ---- Final scaling pass: out = A * rowScale[r] * colScale[j] -------------
  for (int r = wave; r < SK_N; r += 16) {
    const float rs  = rowScale[r];
    const float* Ar = A + (size_t)r * SK_N;
    float*       Or = O + (size_t)r * SK_N;
    #pragma unroll
    for (int k = 0; k < SK_N / 128; ++k) {
      const int j = lane * 4 + k * 128;
      const float4 av = *(const float4*)(Ar + j);
      const float4 cv = *(const float4*)(&colScale[j]);
      float4 ov;
      ov.x = av.x * rs * cv.x;
      ov.y = av.y * rs * cv.y;
      ov.z = av.z * rs * cv.z;
      ov.w = av.w * rs * cv.w;
      *(float4*)(Or + j) = ov;
    }
  }
}

extern "C" void kernel_launch(void* const* d_in, const int* in_sizes, int n_in,
                              void* d_out, int out_size, void* d_ws, size_t ws_size,
                              hipStream_t stream) {
  const float* s = (const float*)d_in[0];
  float* o       = (float*)d_out;
  const int batches = in_sizes[0] / (SK_N * SK_N);   // 256 for the reference
  sinkhorn512_wmma<<<dim3(batches), dim3(512), 0, stream>>>(s, o);
}